// AttentionNet_4698694222022
// MI455X (gfx1250) — compile-verified
//
#include <hip/hip_runtime.h>
#include <cmath>

#define BATCH 8
#define KK 4
#define IMG 448
#define OUTD 224

typedef float v2f __attribute__((ext_vector_type(2)));
typedef float v8f __attribute__((ext_vector_type(8)));

struct Cfg { int kh, kw, ph, pw, stride, gh, gw, layer, a, A; };

// Exactly replicates anchor_configs(): double math, int() truncation, ceil().
static void build_cfgs(Cfg cfgs[21]) {
  const double ars[3] = {0.667, 1.0, 1.5};
  const double s13 = pow(2.0, 1.0 / 3.0);
  const double s23 = pow(2.0, 2.0 / 3.0);
  const double sc3[3] = {s13, s23, 0.0};
  const double sc5[3] = {1.0, s13, s23};
  const int strides[3] = {32, 64, 128};
  const int sizes[3]   = {48, 96, 192};
  int idx = 0;
  for (int layer = 0; layer < 3; ++layer) {
    const int nsc = (layer == 2) ? 3 : 2;
    const double* sc = (layer == 2) ? sc5 : sc3;
    int a = 0;
    for (int si = 0; si < nsc; ++si) {
      for (int ai = 0; ai < 3; ++ai) {
        const double ss = (double)sizes[layer] * sc[si];
        const double ra = sqrt(ars[ai]);
        Cfg c;
        c.kh = (int)(ss / ra);
        c.kw = (int)(ss * ra);
        c.ph = (int)ceil(((double)c.kh - strides[layer]) / 2.0);
        c.pw = (int)ceil(((double)c.kw - strides[layer]) / 2.0);
        c.stride = strides[layer];
        c.gh = IMG / strides[layer];
        c.gw = c.gh;
        c.layer = layer;
        c.a = a++;
        c.A = (layer == 2) ? 9 : 6;
        cfgs[idx++] = c;
      }
    }
  }
}

// Stage 1: weighted patch aggregation via V_WMMA_F32_16X16X4_F32.
//   D[m=pixel(16), n=k(4 of 16)] += A[m, K=4 g's] * B[K, n]
// A frag (32-bit 16x4): lanes 0-15 hold {P[g0,m],P[g1,m]}, lanes 16-31 {P[g2,m],P[g3,m]}
// B frag (32-bit 4x16): lanes 0-15 hold {W[n,g0],W[n,g1]}, lanes 16-31 {W[n,g2],W[n,g3]}
// Loads are branchless: linear offset clamped into the image, result masked by cndmask.
__global__ __launch_bounds__(256)
void stage1_wmma(const float* __restrict__ x, const float* __restrict__ wts,
                 float* __restrict__ agg,
                 int kh, int kw, int ph, int pw, int stride,
                 int gh, int gw, int A, int a, int bBase, int aggStride)
{
  __shared__ float lds_w[196 * 16];  // [g][16]: cols 0..3 = W[b,a,k,g], rest 0
  const int G  = gh * gw;
  const int Gp = (G + 3) & ~3;
  const int b  = bBase + blockIdx.x;

  for (int t = threadIdx.x; t < Gp * 16; t += 256) {
    const int g = t >> 4, n = t & 15;
    float v = 0.f;
    if (g < G && n < KK) {
      const int i = g / gw, j = g - i * gw;
      v = wts[((((size_t)b * A + a) * KK + n) * gh + i) * gw + j];
    }
    lds_w[t] = v;
  }
  __syncthreads();

  const int lane  = threadIdx.x & 31;
  const int wave  = threadIdx.x >> 5;
  const int npix  = kh * kw * 3;
  const int ntiles = (npix + 15) >> 4;
  const int tile  = blockIdx.y * 8 + wave;
  const bool tileOk = (tile < ntiles);

  const int half2 = (lane >= 16) ? 2 : 0;
  const int n = lane & 15;

  int m = tile * 16 + n;
  const bool mOk = tileOk && (m < npix);
  if (!mOk) m = 0;
  const int hw = m / 3;
  const int c  = m - hw * 3;
  const int h  = hw / kw;
  const int w  = hw - h * kw;
  const int r0 = h - ph;
  const int c0 = w - pw;

  const float* xb = x + (size_t)b * (IMG * IMG * 3);
  __builtin_prefetch(xb + (r0 < 0 ? 0 : r0) * (IMG * 3), 0, 1);  // global_prefetch_b8

  v8f acc = {0.f, 0.f, 0.f, 0.f, 0.f, 0.f, 0.f, 0.f};

  const int maxOff = IMG * IMG * 3 - 1;
  int gi = 0, gj = half2;  // decode of glo = half2 (gw >= 3 always)
  const int nq = Gp >> 2;
  for (int q = 0; q < nq; ++q) {
    const int glo = (q << 2) + half2;
    int j1 = gj + 1, i1 = gi;
    if (j1 >= gw) { j1 = 0; i1 = gi + 1; }

    const int rr0 = gi * stride + r0, cc0 = gj * stride + c0;
    const int rr1 = i1 * stride + r0, cc1 = j1 * stride + c0;

    const bool ok0 = mOk && (glo < G) &&
                     ((unsigned)rr0 < IMG) && ((unsigned)cc0 < IMG);
    const bool ok1 = mOk && ((glo + 1) < G) &&
                     ((unsigned)rr1 < IMG) && ((unsigned)cc1 < IMG);

    // Clamp linear offsets so the loads are always in-bounds (branchless).
    int ad0 = rr0 * (IMG * 3) + cc0 * 3 + c;
    int ad1 = rr1 * (IMG * 3) + cc1 * 3 + c;
    ad0 = ad0 < 0 ? 0 : (ad0 > maxOff ? maxOff : ad0);
    ad1 = ad1 < 0 ? 0 : (ad1 > maxOff ? maxOff : ad1);

    const float v0 = xb[ad0];
    const float v1 = xb[ad1];
    const float a0 = ok0 ? v0 : 0.f;
    const float a1 = ok1 ? v1 : 0.f;

    v2f av = {a0, a1};
    v2f bv = { lds_w[(glo << 4) + n], lds_w[((glo + 1) << 4) + n] };

    acc = __builtin_amdgcn_wmma_f32_16x16x4_f32(false, av, false, bv,
                                                (short)0, acc, false, false);
    gj += 4; if (gj >= gw) { gj -= gw; ++gi; }
  }

  // D layout: VGPR r, lanes 0-15 -> M=r, lanes 16-31 -> M=r+8; N = lane&15 = k
  if (tileOk && n < KK) {
    float* dst = agg + (size_t)(blockIdx.x * KK + n) * aggStride;
    const int p0 = tile * 16 + ((lane >= 16) ? 8 : 0);
    if (p0 + 8 <= npix) {
      *(float4*)(dst + p0)     = make_float4(acc[0], acc[1], acc[2], acc[3]);
      *(float4*)(dst + p0 + 4) = make_float4(acc[4], acc[5], acc[6], acc[7]);
    } else {
#pragma unroll
      for (int r = 0; r < 8; ++r)
        if (p0 + r < npix) dst[p0 + r] = acc[r];
    }
  }
}

// Stage 2: jax.image.resize(..., 'bilinear', antialias=True) semantics + accumulate.
__global__ __launch_bounds__(256)
void stage2_resize_acc(const float* __restrict__ agg, float* __restrict__ out,
                       int kh, int kw, int aggStride, int bkCount, int bkBase)
{
  const int idx = blockIdx.x * 256 + threadIdx.x;
  const int total = bkCount * OUTD * OUTD * 3;
  if (idx >= total) return;
  const int c = idx % 3;
  int t = idx / 3;
  const int ow = t % OUTD; t /= OUTD;
  const int oh = t % OUTD;
  const int bk = t / OUTD;

  // H taps
  const float invh = (float)kh / (float)OUTD;
  const float ksh  = invh > 1.f ? invh : 1.f;
  const float sfh  = (oh + 0.5f) * invh - 0.5f;
  int h0 = (int)ceilf(sfh - ksh);  if (h0 < 0) h0 = 0;
  int h1 = (int)floorf(sfh + ksh); if (h1 > kh - 1) h1 = kh - 1;
  int nh = h1 - h0 + 1; if (nh > 6) nh = 6;
  float whv[6]; float hsum = 0.f;
  for (int i = 0; i < nh; ++i) {
    float d = fabsf(sfh - (float)(h0 + i)) / ksh;
    float wv = 1.f - d; wv = wv > 0.f ? wv : 0.f;
    whv[i] = wv; hsum += wv;
  }
  // W taps
  const float invw = (float)kw / (float)OUTD;
  const float ksw  = invw > 1.f ? invw : 1.f;
  const float sfw  = (ow + 0.5f) * invw - 0.5f;
  int w0 = (int)ceilf(sfw - ksw);  if (w0 < 0) w0 = 0;
  int w1 = (int)floorf(sfw + ksw); if (w1 > kw - 1) w1 = kw - 1;
  int nw = w1 - w0 + 1; if (nw > 6) nw = 6;
  float wwv[6]; float wsum = 0.f;
  for (int j = 0; j < nw; ++j) {
    float d = fabsf(sfw - (float)(w0 + j)) / ksw;
    float wv = 1.f - d; wv = wv > 0.f ? wv : 0.f;
    wwv[j] = wv; wsum += wv;
  }

  const float* src = agg + (size_t)bk * aggStride;
  float s = 0.f;
  for (int i = 0; i < nh; ++i) {
    const float* row = src + (size_t)(h0 + i) * (kw * 3) + w0 * 3 + c;
    float rs = 0.f;
    for (int j = 0; j < nw; ++j) rs += wwv[j] * row[j * 3];
    s += whv[i] * rs;
  }
  const float norm = (hsum > 0.f ? 1.f / hsum : 0.f) * (wsum > 0.f ? 1.f / wsum : 0.f);
  out[(size_t)bkBase * (OUTD * OUTD * 3) + idx] += s * norm;
}

extern "C" void kernel_launch(void* const* d_in, const int* in_sizes, int n_in,
                              void* d_out, int out_size, void* d_ws, size_t ws_size,
                              hipStream_t stream)
{
  const float* x = (const float*)d_in[0];
  const float* wp[3] = {(const float*)d_in[1], (const float*)d_in[2], (const float*)d_in[3]};
  float* out = (float*)d_out;
  float* agg = (float*)d_ws;

  hipMemsetAsync(d_out, 0, (size_t)out_size * sizeof(float), stream);

  Cfg cfgs[21];
  build_cfgs(cfgs);

  // Workspace budget: largest config needs bCount*4*aggStride floats (<= 35.6 MB at bCount=8).
  size_t maxA = 0;
  for (int i = 0; i < 21; ++i) {
    size_t npa = (((size_t)cfgs[i].kh * cfgs[i].kw * 3) + 7) & ~(size_t)7;
    if (npa > maxA) maxA = npa;
  }
  int bCount = BATCH;
  while (bCount > 1 && (size_t)bCount * KK * maxA * sizeof(float) > ws_size) bCount >>= 1;

  for (int ci = 0; ci < 21; ++ci) {
    const Cfg& cf = cfgs[ci];
    const int npix = cf.kh * cf.kw * 3;
    const int aggStride = (npix + 7) & ~7;
    const int ntiles = (npix + 15) >> 4;
    const int tileBlocks = (ntiles + 7) >> 3;
    for (int bBase = 0; bBase < BATCH; bBase += bCount) {
      const int bThis = (BATCH - bBase < bCount) ? (BATCH - bBase) : bCount;
      dim3 g1(bThis, tileBlocks);
      stage1_wmma<<<g1, 256, 0, stream>>>(x, wp[cf.layer], agg,
          cf.kh, cf.kw, cf.ph, cf.pw, cf.stride, cf.gh, cf.gw,
          cf.A, cf.a, bBase, aggStride);
      const int bkCount = bThis * KK;
      const int total = bkCount * OUTD * OUTD * 3;
      const int blocks = (total + 255) / 256;
      stage2_resize_acc<<<blocks, 256, 0, stream>>>(agg, out,
          cf.kh, cf.kw, aggStride, bkCount, bBase * KK);
    }
  }
}